// BEVGenerator_80882824119006
// MI455X (gfx1250) — compile-verified
//
#include <hip/hip_runtime.h>
#include <math.h>

// ---------------- problem constants (match reference) ----------------
#define B_BATCH     32
#define N_PTS       131072
#define NUM_SLICES  6
#define HGT         256
#define WID         256
#define EPSV        1e-6f

// ---------------- tiling for the histogram kernel ----------------
#define HIST_THREADS     256          // 8 wave32 per block
#define TILE_PTS         1024         // points per async LDS tile (12 KB)
#define TILES_PER_CHUNK  4
#define CHUNKS           32           // blocks per batch; 32*4*1024 = 131072

// order-preserving float<->uint encoding (for atomic min/max on floats)
__device__ __forceinline__ unsigned f2ord(float f) {
    unsigned u = __float_as_uint(f);
    return (u & 0x80000000u) ? ~u : (u | 0x80000000u);
}
__device__ __forceinline__ float ord2f(unsigned k) {
    unsigned u = (k & 0x80000000u) ? (k & 0x7FFFFFFFu) : ~k;
    return __uint_as_float(u);
}

// wave32 butterfly min/max (cross-lane VALU, no LDS, no barriers)
__device__ __forceinline__ float wave_min(float v) {
#pragma unroll
    for (int off = 16; off > 0; off >>= 1)
        v = fminf(v, __shfl_xor(v, off, 32));
    return v;
}
__device__ __forceinline__ float wave_max(float v) {
#pragma unroll
    for (int off = 16; off > 0; off >>= 1)
        v = fmaxf(v, __shfl_xor(v, off, 32));
    return v;
}

// ---------------------------------------------------------------------
// Kernel 1: zero output histogram (b128 stores), init z min/max slots.
// out_size = 12,582,912 floats = 3,145,728 float4.
// ws layout: ws[0..31] = enc zmin (init 0xFFFFFFFF), ws[32..63] = enc zmax (init 0)
// ---------------------------------------------------------------------
__global__ void __launch_bounds__(1024)
init_kernel(float4* __restrict__ out4, unsigned n4, unsigned* __restrict__ zmm) {
    unsigned i = blockIdx.x * blockDim.x + threadIdx.x;
    if (i < n4) out4[i] = make_float4(0.f, 0.f, 0.f, 0.f);
    if (blockIdx.x == 0 && threadIdx.x < 2u * B_BATCH)
        zmm[threadIdx.x] = (threadIdx.x < B_BATCH) ? 0xFFFFFFFFu : 0u;
}

// ---------------------------------------------------------------------
// Kernel 2: per-batch z min/max. 4 points = 3 x float4 (48B, 16B aligned).
// regs -> wave32 shuffle -> LDS across 8 waves -> one atomic per block.
// ---------------------------------------------------------------------
__global__ void __launch_bounds__(256)
zminmax_kernel(const float* __restrict__ xyz, unsigned* __restrict__ zmm) {
    const int b     = blockIdx.y;
    const int chunk = blockIdx.x;          // 8 chunks per batch
    const int tid   = threadIdx.x;         // 256 threads
    const int quads = (N_PTS / 8) / 4;     // 4096 point-quads per block
    const float4* base = (const float4*)(xyz + ((size_t)b * N_PTS +
                                                (size_t)chunk * (N_PTS / 8)) * 3);

    float mn = __builtin_inff(), mx = -__builtin_inff();
    for (int q = tid; q < quads; q += 256) {
        float4 a = base[3 * q + 0];        // p0.x p0.y p0.z p1.x
        float4 c = base[3 * q + 1];        // p1.y p1.z p2.x p2.y
        float4 d = base[3 * q + 2];        // p2.z p3.x p3.y p3.z
        float z0 = a.z, z1 = c.y, z2 = d.x, z3 = d.w;
        mn = fminf(fminf(fminf(mn, z0), fminf(z1, z2)), z3);
        mx = fmaxf(fmaxf(fmaxf(mx, z0), fmaxf(z1, z2)), z3);
    }
    mn = wave_min(mn);
    mx = wave_max(mx);

    __shared__ float smn[8], smx[8];
    const int lane = tid & 31, wave = tid >> 5;
    if (lane == 0) { smn[wave] = mn; smx[wave] = mx; }
    __syncthreads();
    if (wave == 0) {
        float m0 = (lane < 8) ? smn[lane] :  __builtin_inff();
        float m1 = (lane < 8) ? smx[lane] : -__builtin_inff();
        m0 = wave_min(m0);
        m1 = wave_max(m1);
        if (lane == 0) {
            atomicMin(&zmm[b],           f2ord(m0));
            atomicMax(&zmm[B_BATCH + b], f2ord(m1));
        }
    }
}

// ---------------------------------------------------------------------
// Async global->LDS tile copy (CDNA5 async path, ASYNCcnt-tracked).
// Copies TILE_PTS*12 = 12288 bytes: 3 x b128 per thread (256 threads).
// GVS addressing: mem = SGPR64(base) + VGPR32(offset); VDST = LDS byte addr.
// ---------------------------------------------------------------------
__device__ __forceinline__ void async_tile_load(const float* gbase,
                                                unsigned lds_base, int tid) {
    unsigned voff = (unsigned)tid * 16u;
#pragma unroll
    for (int i = 0; i < 3; ++i) {
        asm volatile("global_load_async_to_lds_b128 %0, %1, %2"
                     :: "v"(lds_base + voff), "v"(voff), "s"(gbase)
                     : "memory");
        voff += HIST_THREADS * 16u;
    }
}

// ---------------------------------------------------------------------
// Kernel 3: histogram scatter. Double-buffered async LDS staging of xyz.
// ---------------------------------------------------------------------
__global__ void __launch_bounds__(HIST_THREADS)
hist_kernel(const float* __restrict__ xyz, const unsigned* __restrict__ zmm,
            float* __restrict__ out) {
    const int b     = blockIdx.y;
    const int chunk = blockIdx.x;
    const int tid   = threadIdx.x;

    __shared__ __align__(16) float buf[2][TILE_PTS * 3];

    const float zmin = ord2f(zmm[b]);
    const float zmax = ord2f(zmm[B_BATCH + b]);
    float edge[NUM_SLICES + 1];
#pragma unroll
    for (int k = 0; k <= NUM_SLICES; ++k)
        edge[k] = zmin + (zmax - zmin) * ((float)k / (float)NUM_SLICES);

    const float* gbase =
        xyz + ((size_t)b * N_PTS + (size_t)chunk * (TILE_PTS * TILES_PER_CHUNK)) * 3;
    float* outb = out + (size_t)b * NUM_SLICES * HGT * WID;

    // prologue: fill buffer 0
    async_tile_load(gbase, (unsigned)(size_t)&buf[0][0], tid);

    for (int t = 0; t < TILES_PER_CHUNK; ++t) {
        // prefetch next tile into the other buffer while we process this one
        if (t + 1 < TILES_PER_CHUNK) {
            async_tile_load(gbase + (size_t)(t + 1) * TILE_PTS * 3,
                            (unsigned)(size_t)&buf[(t + 1) & 1][0], tid);
            // 3 outstanding async ops belong to the prefetch; wait for tile t
            asm volatile("s_wait_asynccnt 3" ::: "memory");
        } else {
            asm volatile("s_wait_asynccnt 0" ::: "memory");
        }
        __syncthreads();   // publish LDS tile across waves

        const float* sb = &buf[t & 1][0];
#pragma unroll
        for (int p = 0; p < TILE_PTS / HIST_THREADS; ++p) {
            int   idx = tid + p * HIST_THREADS;
            float x = sb[3 * idx + 0];
            float y = sb[3 * idx + 1];
            float z = sb[3 * idx + 2];
            // gx = (x - X_MIN)/(X_MAX - X_MIN + EPS)*(W-1), same op order as ref
            float gx = (x + 1.0f) / (2.0f + EPSV) * (float)(WID - 1);
            float gy = (y + 1.0f) / (2.0f + EPSV) * (float)(HGT - 1);
            bool valid = (gy >= 0.0f) & (gy < (float)HGT) &
                         (gx >= 0.0f) & (gx < (float)WID);
            int iy = min(max((int)gy, 0), HGT - 1);
            int ix = min(max((int)gx, 0), WID - 1);
            int s = -1;
#pragma unroll
            for (int k = 0; k < NUM_SLICES; ++k)
                if (z >= edge[k] && z < edge[k + 1]) s = k;
            if (valid && s >= 0)
                atomicAdd(&outb[((size_t)s * HGT + iy) * WID + ix], 1.0f);
        }
        __syncthreads();   // all waves done reading before buffer is refilled
    }
}

// ---------------------------------------------------------------------
// Kernel 4: per-(b,s) plane: in-place log1p, min/max, normalize.
// One 1024-thread block per plane; float4 traffic; wave32 shuffle reduce.
// ---------------------------------------------------------------------
__global__ void __launch_bounds__(1024)
finalize_kernel(float* __restrict__ out) {
    const int plane = blockIdx.x;          // B*S = 192
    const int tid   = threadIdx.x;         // 1024 = 32 waves
    float4* p4 = (float4*)(out + (size_t)plane * (HGT * WID));
    const int n4 = (HGT * WID) / 4;        // 16384

    float mn = __builtin_inff(), mx = -__builtin_inff();
    for (int i = tid; i < n4; i += 1024) {
        float4 v = p4[i];
        v.x = log1pf(v.x); v.y = log1pf(v.y);
        v.z = log1pf(v.z); v.w = log1pf(v.w);
        p4[i] = v;
        mn = fminf(mn, fminf(fminf(v.x, v.y), fminf(v.z, v.w)));
        mx = fmaxf(mx, fmaxf(fmaxf(v.x, v.y), fmaxf(v.z, v.w)));
    }
    mn = wave_min(mn);
    mx = wave_max(mx);

    __shared__ float smn[32], smx[32];
    const int lane = tid & 31, wave = tid >> 5;
    if (lane == 0) { smn[wave] = mn; smx[wave] = mx; }
    __syncthreads();
    // all waves redundantly reduce the 32 partials (broadcast-free)
    float lo = wave_min(smn[lane]);
    float hi = wave_max(smx[lane]);
    float inv = 1.0f / (hi - lo + EPSV);

    for (int i = tid; i < n4; i += 1024) {
        float4 v = p4[i];
        v.x = (v.x - lo) * inv; v.y = (v.y - lo) * inv;
        v.z = (v.z - lo) * inv; v.w = (v.w - lo) * inv;
        p4[i] = v;
    }
}

// ---------------------------------------------------------------------
extern "C" void kernel_launch(void* const* d_in, const int* in_sizes, int n_in,
                              void* d_out, int out_size, void* d_ws, size_t ws_size,
                              hipStream_t stream) {
    (void)in_sizes; (void)n_in; (void)ws_size;
    const float* xyz = (const float*)d_in[0];
    float*       out = (float*)d_out;
    unsigned*    zmm = (unsigned*)d_ws;      // 64 uints used

    unsigned n4 = (unsigned)out_size / 4u;   // 3,145,728 float4
    int init_blocks = (int)((n4 + 1023u) / 1024u);

    init_kernel    <<<init_blocks,              1024, 0, stream>>>((float4*)out, n4, zmm);
    zminmax_kernel <<<dim3(8, B_BATCH),          256, 0, stream>>>(xyz, zmm);
    hist_kernel    <<<dim3(CHUNKS, B_BATCH), HIST_THREADS, 0, stream>>>(xyz, zmm, out);
    finalize_kernel<<<B_BATCH * NUM_SLICES,     1024, 0, stream>>>(out);
}